// FillSimNet_14147622273759
// MI455X (gfx1250) — compile-verified
//
#include <hip/hip_runtime.h>

#define NN 100000
#define NE 1600000

typedef __attribute__((ext_vector_type(2))) float v2f;
typedef __attribute__((ext_vector_type(8))) float v8f;

__device__ __forceinline__ v8f wmma_f32_k4(v2f a, v2f b, v8f c) {
  // D = A(16x4,f32) x B(4x16,f32) + C(16x16,f32)
  return __builtin_amdgcn_wmma_f32_16x16x4_f32(
      /*neg_a=*/false, a, /*neg_b=*/false, b,
      /*c_mod=*/(short)0, c, /*reuse_a=*/false, /*reuse_b=*/false);
}

// Pair-packed, bank-swizzled LDS index for weights:
// element W[k][n] lives at kp*128 + ((2n) ^ ((kp&1)<<5)) + (k&1), kp = k>>1.
// Read side: hi == kp&1, so the two half-waves hit disjoint bank halves and
// each B operand is one aligned 8-byte load into an even VGPR pair.
__device__ __forceinline__ int wpack_idx(int k, int n) {
  int kp = k >> 1;
  return kp * 128 + (((n << 1) ^ ((kp & 1) << 5)) | (k & 1));
}

// Per-wave 16x64 GEMM tile. arow = LDS A row base (row stride applied by
// caller), sWp = pair-packed weights. Accumulates into acc[4] (4 n-tiles).
template <int K>
__device__ __forceinline__ void gemm_tile(const float* arow, const float* sWp,
                                          int l15, int hi, v8f acc[4]) {
#pragma unroll
  for (int k0 = 0; k0 < K; k0 += 4) {
    const int ka = k0 + (hi << 1);
    v2f a = *(const v2f*)(arow + ka);
    const float* bp = sWp + (ka >> 1) * 128;
    const int sw = hi << 5;
#pragma unroll
    for (int nt = 0; nt < 4; ++nt) {
      const int nn = nt * 16 + l15;
      v2f b = *(const v2f*)(bp + ((nn << 1) ^ sw));
      acc[nt] = wmma_f32_k4(a, b, acc[nt]);
    }
  }
}

// Cooperative staging of a KxN=Kx64 weight matrix into packed LDS layout.
template <int K>
__device__ __forceinline__ void stage_weights(const float* __restrict__ gW,
                                              float* sWp, int tid) {
  for (int i = tid; i < K * 64; i += 256) {
    sWp[wpack_idx(i >> 6, i & 63)] = gW[i];
  }
}

// ---------------- degree / normalization ----------------

__global__ void k_deg_init(float* deg) {
  int i = blockIdx.x * 256 + threadIdx.x;
  if (i < NN) deg[i] = 1.0f;  // self-loop weight
}

__global__ void k_deg_accum(const int* __restrict__ dst,
                            const float* __restrict__ w, float* deg) {
  int e = blockIdx.x * 256 + threadIdx.x;
  if (e < NE) atomicAdd(&deg[dst[e]], w[e]);
}

__global__ void k_dinv(const float* __restrict__ deg, float* __restrict__ dinv) {
  int i = blockIdx.x * 256 + threadIdx.x;
  if (i < NN) {
    float d = deg[i];
    dinv[i] = d > 0.f ? rsqrtf(d) : 0.f;
  }
}

__global__ void k_norm(const int* __restrict__ src, const int* __restrict__ dst,
                       const float* __restrict__ w, const float* __restrict__ dinv,
                       float* __restrict__ nrm) {
  int e = blockIdx.x * 256 + threadIdx.x;
  if (e < NE) nrm[e] = dinv[src[e]] * w[e] * dinv[dst[e]];
}

// ---------------- fused GCN linear: t = act(in + pre_b) @ W ----------------
// Also writes self-loop init: agg[i] = dinv[i]^2 * t[i].
// `in` may alias `t_out` (tile fully staged to LDS before stores).

__launch_bounds__(256) __global__
void k_lin64(const float* in, const float* __restrict__ gW,
             const float* __restrict__ pre_bias, int pre_relu,
             float* t_out, float* __restrict__ agg_out,
             const float* __restrict__ dinv) {
  __shared__ float sA[128 * 68];
  __shared__ float sW[32 * 128];
  const int tid = threadIdx.x;
  const int base = blockIdx.x * 128;

  stage_weights<64>(gW, sW, tid);
  // stage A tile (128x64), apply deferred bias+relu, row stride 68
  for (int i = tid; i < 2048; i += 256) {
    int r = i >> 4;
    int c4 = (i & 15) << 2;
    int gr = base + r;
    float4 v = make_float4(0.f, 0.f, 0.f, 0.f);
    if (gr < NN) v = ((const float4*)(in + (size_t)gr * 64))[i & 15];
    if (pre_bias) {
      v.x += pre_bias[c4 + 0]; v.y += pre_bias[c4 + 1];
      v.z += pre_bias[c4 + 2]; v.w += pre_bias[c4 + 3];
    }
    if (pre_relu) {
      v.x = fmaxf(v.x, 0.f); v.y = fmaxf(v.y, 0.f);
      v.z = fmaxf(v.z, 0.f); v.w = fmaxf(v.w, 0.f);
    }
    *(float4*)(sA + r * 68 + c4) = v;  // 16B-aligned: 68r+c4 ≡ 0 (mod 4)
  }
  __syncthreads();

  const int lane = tid & 31, l15 = lane & 15, hi = lane >> 4;
  const int wr0 = (tid >> 5) << 4;
  v8f acc[4] = {{}, {}, {}, {}};
  gemm_tile<64>(sA + (wr0 + l15) * 68, sW, l15, hi, acc);

#pragma unroll
  for (int j = 0; j < 8; ++j) {
    int gr = base + wr0 + j + (hi << 3);
    if (gr < NN) {
      float di = dinv[gr];
      float d2 = di * di;
      size_t ro = (size_t)gr * 64 + l15;
#pragma unroll
      for (int nt = 0; nt < 4; ++nt) {
        float v = acc[nt][j];
        t_out[ro + nt * 16] = v;
        agg_out[ro + nt * 16] = d2 * v;  // self-loop term
      }
    }
  }
}

// ---------------- edge scatter: agg[dst] += norm * t[src] ----------------

__launch_bounds__(256) __global__
void k_scatter(const int* __restrict__ src, const int* __restrict__ dst,
               const float* __restrict__ nrm, const float* __restrict__ tbuf,
               float* agg) {
  long long idx = (long long)blockIdx.x * 256 + threadIdx.x;
  int e = (int)(idx >> 6);
  int c = (int)(idx & 63);
  if (e < NE) {
    float msg = nrm[e] * tbuf[(size_t)src[e] * 64 + c];
    atomicAdd(&agg[(size_t)dst[e] * 64 + c], msg);
  }
}

// ---------------- fused encoder: h = relu(x@W1+b1)@W2 + b2 ----------------

__launch_bounds__(256) __global__
void k_encoder(const float* __restrict__ x, const float* __restrict__ w1,
               const float* __restrict__ b1, const float* __restrict__ w2,
               const float* __restrict__ b2, float* __restrict__ hout) {
  __shared__ float sX[128 * 12];
  __shared__ float sW1[4 * 128];
  __shared__ float sH[128 * 68];
  __shared__ float sW2[32 * 128];
  const int tid = threadIdx.x;
  const int base = blockIdx.x * 128;

  stage_weights<8>(w1, sW1, tid);
  stage_weights<64>(w2, sW2, tid);
  for (int i = tid; i < 256; i += 256) {  // 128 rows x 8 cols
    int r = i >> 1, c4 = (i & 1) << 2;
    int gr = base + r;
    float4 v = make_float4(0.f, 0.f, 0.f, 0.f);
    if (gr < NN) v = ((const float4*)(x + (size_t)gr * 8))[i & 1];
    *(float4*)(sX + r * 12 + c4) = v;
  }
  __syncthreads();

  const int lane = tid & 31, l15 = lane & 15, hi = lane >> 4;
  const int wr0 = (tid >> 5) << 4;

  v8f acc[4] = {{}, {}, {}, {}};
  gemm_tile<8>(sX + (wr0 + l15) * 12, sW1, l15, hi, acc);  // GEMM1: K=8

#pragma unroll
  for (int j = 0; j < 8; ++j) {  // h1 = relu(. + b1) -> LDS
    int r = wr0 + j + (hi << 3);
#pragma unroll
    for (int nt = 0; nt < 4; ++nt)
      sH[r * 68 + nt * 16 + l15] = fmaxf(acc[nt][j] + b1[nt * 16 + l15], 0.f);
  }
  __syncthreads();

  v8f acc2[4] = {{}, {}, {}, {}};
  gemm_tile<64>(sH + (wr0 + l15) * 68, sW2, l15, hi, acc2);  // GEMM2

#pragma unroll
  for (int j = 0; j < 8; ++j) {
    int gr = base + wr0 + j + (hi << 3);
    if (gr < NN) {
      size_t ro = (size_t)gr * 64 + l15;
#pragma unroll
      for (int nt = 0; nt < 4; ++nt)
        hout[ro + nt * 16] = acc2[nt][j] + b2[nt * 16 + l15];
    }
  }
}

// --------- fused decoder: out = relu(relu(h_raw+gb)@W1+b1) @ w2 + b2 ---------

__launch_bounds__(256) __global__
void k_decoder(const float* __restrict__ hin, const float* __restrict__ gcn_b3,
               const float* __restrict__ w1, const float* __restrict__ b1,
               const float* __restrict__ w2, const float* __restrict__ b2,
               float* __restrict__ out) {
  __shared__ float sA[128 * 68];
  __shared__ float sW[32 * 128];
  __shared__ float sw2[64];
  const int tid = threadIdx.x;
  const int base = blockIdx.x * 128;

  if (tid < 64) sw2[tid] = w2[tid];
  stage_weights<64>(w1, sW, tid);
  for (int i = tid; i < 2048; i += 256) {  // stage h with deferred bias+relu
    int r = i >> 4;
    int c4 = (i & 15) << 2;
    int gr = base + r;
    float4 v = make_float4(0.f, 0.f, 0.f, 0.f);
    if (gr < NN) v = ((const float4*)(hin + (size_t)gr * 64))[i & 15];
    v.x = fmaxf(v.x + gcn_b3[c4 + 0], 0.f);
    v.y = fmaxf(v.y + gcn_b3[c4 + 1], 0.f);
    v.z = fmaxf(v.z + gcn_b3[c4 + 2], 0.f);
    v.w = fmaxf(v.w + gcn_b3[c4 + 3], 0.f);
    *(float4*)(sA + r * 68 + c4) = v;
  }
  __syncthreads();

  const int lane = tid & 31, l15 = lane & 15, hi = lane >> 4;
  const int wr0 = (tid >> 5) << 4;
  v8f acc[4] = {{}, {}, {}, {}};
  gemm_tile<64>(sA + (wr0 + l15) * 68, sW, l15, hi, acc);

#pragma unroll
  for (int j = 0; j < 8; ++j) {  // d1 = relu(. + b1) back into own rows of sA
    int r = wr0 + j + (hi << 3);
#pragma unroll
    for (int nt = 0; nt < 4; ++nt)
      sA[r * 68 + nt * 16 + l15] = fmaxf(acc[nt][j] + b1[nt * 16 + l15], 0.f);
  }
  __syncthreads();

  if (tid < 128) {  // final [64]x[64,1] dot per row
    int gr = base + tid;
    if (gr < NN) {
      float s = b2[0];
      const float* row = sA + tid * 68;
#pragma unroll
      for (int c = 0; c < 64; ++c) s += row[c] * sw2[c];
      out[gr] = s;
    }
  }
}

// ---------------- host launch ----------------

extern "C" void kernel_launch(void* const* d_in, const int* in_sizes, int n_in,
                              void* d_out, int out_size, void* d_ws, size_t ws_size,
                              hipStream_t stream) {
  const float* x      = (const float*)d_in[0];
  const int*   ei     = (const int*)d_in[1];
  const float* ew     = (const float*)d_in[2];
  const float* enc_w1 = (const float*)d_in[3];
  const float* enc_b1 = (const float*)d_in[4];
  const float* enc_w2 = (const float*)d_in[5];
  const float* enc_b2 = (const float*)d_in[6];
  const float* gcn_w  = (const float*)d_in[7];
  const float* gcn_b  = (const float*)d_in[8];
  const float* dec_w1 = (const float*)d_in[9];
  const float* dec_b1 = (const float*)d_in[10];
  const float* dec_w2 = (const float*)d_in[11];
  const float* dec_b2 = (const float*)d_in[12];
  float* out = (float*)d_out;

  const int* src = ei;        // edge_index[0]
  const int* dst = ei + NE;   // edge_index[1]

  float* ws = (float*)d_ws;
  const size_t N64 = (size_t)NN * 64;
  float* H    = ws;             // node features (ping)
  float* G    = ws + N64;       // node features (pong)
  float* deg  = ws + 2 * N64;   // [NN]
  float* dinv = deg + NN;       // [NN]
  float* nrm  = dinv + NN;      // [NE]

  const dim3 blk(256);
  const int nbN = (NN + 255) / 256;
  const int nbE = (NE + 255) / 256;
  const int nbL = (NN + 127) / 128;
  const int nbS = (int)(((long long)NE * 64 + 255) / 256);

  k_deg_init<<<nbN, blk, 0, stream>>>(deg);
  k_deg_accum<<<nbE, blk, 0, stream>>>(dst, ew, deg);
  k_dinv<<<nbN, blk, 0, stream>>>(deg, dinv);
  k_norm<<<nbE, blk, 0, stream>>>(src, dst, ew, dinv, nrm);

  k_encoder<<<nbL, blk, 0, stream>>>(x, enc_w1, enc_b1, enc_w2, enc_b2, H);

  // layer 0: in=H (h0 final), t in-place in H, agg -> G
  k_lin64<<<nbL, blk, 0, stream>>>(H, gcn_w + 0 * 4096, nullptr, 0, H, G, dinv);
  k_scatter<<<nbS, blk, 0, stream>>>(src, dst, nrm, H, G);
  // layer 1: in=G raw agg0, apply b0+relu on load; t->G, agg -> H
  k_lin64<<<nbL, blk, 0, stream>>>(G, gcn_w + 1 * 4096, gcn_b + 0 * 64, 1, G, H, dinv);
  k_scatter<<<nbS, blk, 0, stream>>>(src, dst, nrm, G, H);
  // layer 2
  k_lin64<<<nbL, blk, 0, stream>>>(H, gcn_w + 2 * 4096, gcn_b + 1 * 64, 1, H, G, dinv);
  k_scatter<<<nbS, blk, 0, stream>>>(src, dst, nrm, H, G);
  // layer 3
  k_lin64<<<nbL, blk, 0, stream>>>(G, gcn_w + 3 * 4096, gcn_b + 2 * 64, 1, G, H, dinv);
  k_scatter<<<nbS, blk, 0, stream>>>(src, dst, nrm, G, H);

  // decoder reads raw agg3 in H, applies gcn_b[3]+relu on load
  k_decoder<<<nbL, blk, 0, stream>>>(H, gcn_b + 3 * 64, dec_w1, dec_b1, dec_w2,
                                     dec_b2, out);
}